// LocalGrouperKnn_12506944766652
// MI455X (gfx1250) — compile-verified
//
#include <hip/hip_runtime.h>
#include <hip/hip_bf16.h>

#define DEV __device__ __forceinline__

typedef float v2f __attribute__((ext_vector_type(2)));
typedef float v8f __attribute__((ext_vector_type(8)));

static constexpr int B = 16, N = 4096, S = 1024;
static constexpr float EPS = 1e-5f;

static constexpr int P0 = B * S * 32;   // branch0 positions (524288)
static constexpr int P1 = B * S * 64;   // branch1 positions (1048576)

// ---- workspace layout (element offsets; idx region is int, rest float) ----
static constexpr size_t WS_DIFF1 = (size_t)B * S * 64;          // after idx[int B*S*64]
static constexpr size_t WS_DIFF0 = WS_DIFF1 + 3 * (size_t)P1;   // diff1: 3*P1 floats
static constexpr size_t WS_SMALL = WS_DIFF0 + 3 * (size_t)P0;   // diff0: 3*P0 floats
static constexpr size_t WS_SS    = WS_SMALL;                    // std sums [2][16][2]
static constexpr size_t WS_SI    = WS_SS + 64;                  // stdinv  [2][16]
static constexpr size_t WS_ST    = WS_SI + 32;                  // BN stat accums (576)
static constexpr size_t WS_BN    = WS_ST + 576;                 // BN a/c coeffs (576)
static constexpr int    SMALL_CNT = 64 + 32 + 576 + 576;

DEV v8f wmma4(v2f a, v2f b, v8f c) {
  // V_WMMA_F32_16X16X4_F32 : D = A(16x4) * B(4x16) + C, full f32
  return __builtin_amdgcn_wmma_f32_16x16x4_f32(false, a, false, b, (short)0, c,
                                               false, false);
}

// -------------------- init: zero accumulator region --------------------
__global__ void lgk_init_kernel(float* ws) {
  int i = blockIdx.x * 256 + threadIdx.x;
  if (i < SMALL_CNT) ws[WS_SMALL + i] = 0.0f;
}

// -------------------- output part 1: transpose new_xyz --------------------
__global__ __launch_bounds__(256) void lgk_out0_kernel(const float* __restrict__ nxyz,
                                                       float* __restrict__ out) {
  int i = blockIdx.x * 256 + threadIdx.x;           // over B*3*S
  int b = i / (3 * S);
  int r = i - b * 3 * S;
  int c = r / S;
  int s = r - c * S;
  out[i] = nxyz[(size_t)(b * S + s) * 3 + c];
}

// -------------------- KNN: one wave per query, top-64 sorted --------------------
__global__ __launch_bounds__(128) void lgk_knn_kernel(const float* __restrict__ xyz,
                                                      const float* __restrict__ nxyz,
                                                      int* __restrict__ idx) {
  __shared__ float s_d[4][64];
  __shared__ int   s_i[4][64];
  const int wave = threadIdx.x >> 5;
  const int L = threadIdx.x & 31;
  const int g = blockIdx.x * 4 + wave;     // query id in [0, B*S)
  const int b = g >> 10;
  volatile float* ld = s_d[wave];
  volatile int*   li = s_i[wave];
  const float* xb = xyz + (size_t)b * 3 * N;
  const float qx = nxyz[(size_t)g * 3 + 0];
  const float qy = nxyz[(size_t)g * 3 + 1];
  const float qz = nxyz[(size_t)g * 3 + 2];

  auto dist = [&](int n) {
    float dx = xb[n] - qx, dy = xb[N + n] - qy, dz = xb[2 * N + n] - qz;
    return dx * dx + dy * dy + dz * dz;
  };

  // seed list with first 64 candidates
  ld[L] = dist(L);           li[L] = L;
  ld[L + 32] = dist(L + 32); li[L + 32] = L + 32;
  __threadfence_block();

  float curmax; int curslot;
  auto recompute = [&]() {
    float a0 = ld[L], a1 = ld[L + 32];
    float m; int sl;
    if (a0 >= a1) { m = a0; sl = L; } else { m = a1; sl = L + 32; }
    for (int off = 16; off; off >>= 1) {
      float om = __shfl_xor(m, off);
      int   os = __shfl_xor(sl, off);
      if (om > m) { m = om; sl = os; }
    }
    curmax = __shfl(m, 0); curslot = __shfl(sl, 0);
  };
  recompute();

  for (int n0 = 64; n0 < N; n0 += 32) {
    float d = dist(n0 + L);
    unsigned long long mask = __ballot(d < curmax);
    while (mask) {
      int j = __ffsll(mask) - 1;
      mask &= mask - 1;
      float dj = __shfl(d, j);
      if (dj < curmax) {                       // uniform across wave
        if (L == 0) { ld[curslot] = dj; li[curslot] = n0 + j; }
        __threadfence_block();
        recompute();
      }
    }
  }

  // bitonic sort 64 ascending (entries 0..31 = 32-NN set)
  for (int k = 2; k <= 64; k <<= 1) {
    for (int j = k >> 1; j > 0; j >>= 1) {
      #pragma unroll
      for (int t = 0; t < 2; ++t) {
        int i = L + t * 32;
        int l = i ^ j;
        if (l > i) {
          float di = ld[i], dl = ld[l];
          bool up = ((i & k) == 0);
          if ((di > dl) == up) {
            int ii = li[i], il = li[l];
            ld[i] = dl; ld[l] = di;
            li[i] = il; li[l] = ii;
          }
        }
      }
      __threadfence_block();
    }
  }
  idx[(size_t)g * 64 + L] = li[L];
  idx[(size_t)g * 64 + L + 32] = li[L + 32];
}

// -------------------- diff = p - 2q, channel-major; std sums --------------------
__global__ __launch_bounds__(256) void lgk_diff_kernel(const float* __restrict__ xyz,
                                                       const float* __restrict__ nxyz,
                                                       const int* __restrict__ idx,
                                                       float* __restrict__ diff1,
                                                       float* __restrict__ diff0,
                                                       float* __restrict__ stdsum) {
  const int i = blockIdx.x * 256 + threadIdx.x;   // over B*S*64
  const int g = i >> 6, k = i & 63;
  const int b = g >> 10;
  const int id = idx[i];
  const float* xb = xyz + (size_t)b * 3 * N;
  float sA = 0.f, qA = 0.f;
  #pragma unroll
  for (int c = 0; c < 3; ++c) {
    float d = xb[c * N + id] - 2.0f * nxyz[(size_t)g * 3 + c];
    diff1[(size_t)c * P1 + i] = d;
    if (k < 32) diff0[(size_t)c * P0 + (size_t)g * 32 + k] = d;
    sA += d; qA += d * d;
  }
  float s0 = (k < 32) ? sA : 0.f;
  float q0 = (k < 32) ? qA : 0.f;
  for (int off = 16; off; off >>= 1) {
    sA += __shfl_xor(sA, off); qA += __shfl_xor(qA, off);
    s0 += __shfl_xor(s0, off); q0 += __shfl_xor(q0, off);
  }
  if ((threadIdx.x & 31) == 0) {
    atomicAdd(&stdsum[(0 * 16 + b) * 2 + 0], s0);
    atomicAdd(&stdsum[(0 * 16 + b) * 2 + 1], q0);
    atomicAdd(&stdsum[(1 * 16 + b) * 2 + 0], sA);
    atomicAdd(&stdsum[(1 * 16 + b) * 2 + 1], qA);
  }
}

__global__ void lgk_stdfin_kernel(float* ws) {
  int t = threadIdx.x;
  if (t < 32) {
    int br = t >> 4, b = t & 15;
    float n = (float)S * (br ? 64.f : 32.f) * 6.0f;
    float sum6 = 2.f * ws[WS_SS + t * 2 + 0];
    float sq6  = 2.f * ws[WS_SS + t * 2 + 1];
    float mean = sum6 / n;
    float var = fmaxf((sq6 - sum6 * mean) / (n - 1.f), 0.f);  // ddof=1
    ws[WS_SI + br * 16 + b] = 1.0f / (sqrtf(var) + EPS);
  }
}

// ---- layer-1 stats pass: back-to-back WMMAs, register accumulation, one reduction ----
template <int K, int C1>
__global__ __launch_bounds__(256) void lgk_stats1_kernel(const float* __restrict__ diff,
                                                         const float* __restrict__ stdinv,
                                                         const float* __restrict__ w1,
                                                         float* __restrict__ gsum,
                                                         float* __restrict__ gsq, int iters) {
  constexpr int P = B * S * K;
  constexpr int MT1 = C1 / 16;
  const int L = threadIdx.x & 31, n = L & 15, half = L >> 4;
  const int wv = (blockIdx.x * 256 + threadIdx.x) >> 5;
  const int nw = gridDim.x * 8;

  // loop-invariant A fragments: effective weight = w[:, :3] + w[:, 3:]
  v2f aa[MT1];
  #pragma unroll
  for (int mt = 0; mt < MT1; ++mt) {
    const float* wr = w1 + (mt * 16 + n) * 6;
    aa[mt][0] = half ? (wr[2] + wr[5]) : (wr[0] + wr[3]);
    aa[mt][1] = half ? 0.f : (wr[1] + wr[4]);
  }
  float rs[MT1][8], rq[MT1][8];
  #pragma unroll
  for (int mt = 0; mt < MT1; ++mt)
    #pragma unroll
    for (int v = 0; v < 8; ++v) { rs[mt][v] = 0.f; rq[mt][v] = 0.f; }

  for (int it = 0; it < iters; ++it) {
    int t = it * nw + wv;
    int p0 = t * 16;
    float si = stdinv[p0 / (S * K)];
    float x0 = diff[0 * (size_t)P + p0 + n] * si;
    float x1 = diff[1 * (size_t)P + p0 + n] * si;
    float x2 = diff[2 * (size_t)P + p0 + n] * si;
    v2f bb; bb[0] = half ? x2 : x0; bb[1] = half ? 0.f : x1;
    #pragma unroll
    for (int mt = 0; mt < MT1; ++mt) {
      v8f acc = {};
      acc = wmma4(aa[mt], bb, acc);
      #pragma unroll
      for (int v = 0; v < 8; ++v) {
        float val = acc[v];
        rs[mt][v] += val;
        rq[mt][v] += val * val;
      }
    }
  }

  // single cross-lane reduction at the end
  #pragma unroll
  for (int mt = 0; mt < MT1; ++mt)
    #pragma unroll
    for (int v = 0; v < 8; ++v) {
      float sv = rs[mt][v], qv = rq[mt][v];
      for (int off = 8; off; off >>= 1) { sv += __shfl_xor(sv, off); qv += __shfl_xor(qv, off); }
      if (n == 0) {
        int ch = mt * 16 + v + 8 * half;
        atomicAdd(&gsum[ch], sv);
        atomicAdd(&gsq[ch], qv);
      }
    }
}

// -------------------- BN finalize: a=g*rsqrt(var+eps), c=beta-mean*a --------------------
__global__ void lgk_bnfin_kernel(const float* __restrict__ sum, const float* __restrict__ sq,
                                 const float* __restrict__ gam, const float* __restrict__ beta,
                                 float* __restrict__ a, float* __restrict__ c, int C, float n) {
  int t = threadIdx.x;
  if (t < C) {
    float mean = sum[t] / n;
    float var = fmaxf(sq[t] / n - mean * mean, 0.f);   // biased (training-mode BN)
    float ai = gam[t] * rsqrtf(var + EPS);
    a[t] = ai;
    c[t] = beta[t] - mean * ai;
  }
}

// ---- fused MLP pass: GEMM1+BN1+relu -> LDS -> GEMM2 -> (stats2 | BN2+relu+maxpool) ----
template <int K, int C1, int C2, bool FINAL>
__global__ __launch_bounds__(256) void lgk_mlp_kernel(
    const float* __restrict__ diff, const float* __restrict__ stdinv,
    const float* __restrict__ w1, const float* __restrict__ w2,
    const float* __restrict__ a1, const float* __restrict__ c1,
    const float* __restrict__ a2, const float* __restrict__ c2,
    float* __restrict__ gsum, float* __restrict__ gsq,
    float* __restrict__ out, int chOff, int sPerWg) {
  constexpr int KT = K / 16, MT1 = C1 / 16, MT2 = C2 / 16, KC2 = C1 / 4;
  constexpr int P = B * S * K;
  __shared__ float hS[C1][K];
  __shared__ float a1s[C1], c1s[C1];
  for (int c = threadIdx.x; c < C1; c += 256) { a1s[c] = a1[c]; c1s[c] = c1[c]; }
  __syncthreads();
  const int wave = threadIdx.x >> 5, L = threadIdx.x & 31, n = L & 15, half = L >> 4;
  const int mt2 = wave;

  // preload W2 A-fragments for this wave's output-channel tile
  float a0r[KC2], a1r[KC2];
  if (mt2 < MT2) {
    const float* wr = w2 + (size_t)(mt2 * 16 + n) * C1;
    #pragma unroll
    for (int kc = 0; kc < KC2; ++kc) {
      a0r[kc] = wr[4 * kc + (half ? 2 : 0)];
      a1r[kc] = wr[4 * kc + (half ? 3 : 1)];
    }
  }
  float rs[8], rq[8];
  #pragma unroll
  for (int v = 0; v < 8; ++v) { rs[v] = 0.f; rq[v] = 0.f; }

  const int g0 = blockIdx.x * sPerWg;
  for (int sl = 0; sl < sPerWg; ++sl) {
    int g = g0 + sl;
    int b = g >> 10;
    float si = stdinv[b];
    // phase 1: GEMM1 (6->C1 collapsed to K=4 WMMA) + BN1 + relu -> hS
    for (int job = wave; job < KT * MT1; job += 8) {
      int kt = job / MT1, mt = job % MT1;
      int p0 = g * K + kt * 16;
      float x0 = diff[0 * (size_t)P + p0 + n] * si;
      float x1 = diff[1 * (size_t)P + p0 + n] * si;
      float x2 = diff[2 * (size_t)P + p0 + n] * si;
      v2f bb; bb[0] = half ? x2 : x0; bb[1] = half ? 0.f : x1;
      const float* wr = w1 + (mt * 16 + n) * 6;
      v2f aa;
      aa[0] = half ? (wr[2] + wr[5]) : (wr[0] + wr[3]);
      aa[1] = half ? 0.f : (wr[1] + wr[4]);
      v8f acc = {};
      acc = wmma4(aa, bb, acc);
      #pragma unroll
      for (int v = 0; v < 8; ++v) {
        int ch = mt * 16 + v + 8 * half;
        hS[ch][kt * 16 + n] = fmaxf(acc[v] * a1s[ch] + c1s[ch], 0.f);
      }
    }
    __syncthreads();
    // phase 2: GEMM2 (C1->C2) per wave; stats or BN2+relu+maxpool
    float rmax[8];
    #pragma unroll
    for (int v = 0; v < 8; ++v) rmax[v] = -3.0e38f;
    if (mt2 < MT2) {
      for (int kt = 0; kt < KT; ++kt) {
        v8f acc = {};
        #pragma unroll
        for (int kc = 0; kc < KC2; ++kc) {
          v2f aa; aa[0] = a0r[kc]; aa[1] = a1r[kc];
          v2f bb;
          bb[0] = hS[4 * kc + 2 * half][kt * 16 + n];
          bb[1] = hS[4 * kc + 2 * half + 1][kt * 16 + n];
          acc = wmma4(aa, bb, acc);
        }
        #pragma unroll
        for (int v = 0; v < 8; ++v) {
          float val = acc[v];
          if (FINAL) {
            int ch = mt2 * 16 + v + 8 * half;
            val = fmaxf(val * a2[ch] + c2[ch], 0.f);
            rmax[v] = fmaxf(rmax[v], val);
          } else {
            rs[v] += val; rq[v] += val * val;
          }
        }
      }
      if (FINAL) {
        #pragma unroll
        for (int v = 0; v < 8; ++v) {
          float m = rmax[v];
          for (int off = 8; off; off >>= 1) m = fmaxf(m, __shfl_xor(m, off));
          if (n == 0) {
            int ch = chOff + mt2 * 16 + v + 8 * half;
            out[(size_t)b * 192 * S + (size_t)ch * S + (g & 1023)] = m;
          }
        }
      }
    }
    __syncthreads();   // hS reused next s
  }
  if (!FINAL && mt2 < MT2) {
    #pragma unroll
    for (int v = 0; v < 8; ++v) {
      float sv = rs[v], qv = rq[v];
      for (int off = 8; off; off >>= 1) { sv += __shfl_xor(sv, off); qv += __shfl_xor(qv, off); }
      if (n == 0) {
        int ch = mt2 * 16 + v + 8 * half;
        atomicAdd(&gsum[ch], sv);
        atomicAdd(&gsq[ch], qv);
      }
    }
  }
}

extern "C" void kernel_launch(void* const* d_in, const int* in_sizes, int n_in,
                              void* d_out, int out_size, void* d_ws, size_t ws_size,
                              hipStream_t stream) {
  (void)in_sizes; (void)n_in; (void)out_size; (void)ws_size;
  const float* xyz  = (const float*)d_in[0];
  const float* nxyz = (const float*)d_in[2];          // d_in[1] (points) unused by reference
  const float* w0_0 = (const float*)d_in[3];
  const float* g0_0 = (const float*)d_in[5];
  const float* be0_0 = (const float*)d_in[6];
  const float* w0_1 = (const float*)d_in[7];
  const float* g0_1 = (const float*)d_in[9];
  const float* be0_1 = (const float*)d_in[10];
  const float* w1_0 = (const float*)d_in[11];
  const float* g1_0 = (const float*)d_in[13];
  const float* be1_0 = (const float*)d_in[14];
  const float* w1_1 = (const float*)d_in[15];
  const float* g1_1 = (const float*)d_in[17];
  const float* be1_1 = (const float*)d_in[18];

  float* ws = (float*)d_ws;
  int* idx = (int*)d_ws;
  float* diff1 = ws + WS_DIFF1;
  float* diff0 = ws + WS_DIFF0;
  float* stdsum = ws + WS_SS;
  float* stdinv = ws + WS_SI;
  float* st = ws + WS_ST;
  float* sum1b0 = st + 0,   *sq1b0 = st + 32;
  float* sum1b1 = st + 64,  *sq1b1 = st + 128;
  float* sum2b0 = st + 192, *sq2b0 = st + 256;
  float* sum2b1 = st + 320, *sq2b1 = st + 448;
  float* bn = ws + WS_BN;
  float* a1b0 = bn + 0,   *c1b0 = bn + 32;
  float* a1b1 = bn + 64,  *c1b1 = bn + 128;
  float* a2b0 = bn + 192, *c2b0 = bn + 256;
  float* a2b1 = bn + 320, *c2b1 = bn + 448;
  float* out0 = (float*)d_out;
  float* out1 = (float*)d_out + (size_t)B * 3 * S;

  lgk_init_kernel<<<(SMALL_CNT + 255) / 256, 256, 0, stream>>>(ws);
  lgk_out0_kernel<<<B * 3 * S / 256, 256, 0, stream>>>(nxyz, out0);
  lgk_knn_kernel<<<B * S / 4, 128, 0, stream>>>(xyz, nxyz, idx);
  lgk_diff_kernel<<<B * S * 64 / 256, 256, 0, stream>>>(xyz, nxyz, idx, diff1, diff0, stdsum);
  lgk_stdfin_kernel<<<1, 64, 0, stream>>>(ws);

  lgk_stats1_kernel<32, 32><<<64, 256, 0, stream>>>(diff0, stdinv, w0_0, sum1b0, sq1b0,
                                                    (P0 / 16) / 512);
  lgk_stats1_kernel<64, 64><<<64, 256, 0, stream>>>(diff1, stdinv + 16, w1_0, sum1b1, sq1b1,
                                                    (P1 / 16) / 512);
  lgk_bnfin_kernel<<<1, 256, 0, stream>>>(sum1b0, sq1b0, g0_0, be0_0, a1b0, c1b0, 32, (float)P0);
  lgk_bnfin_kernel<<<1, 256, 0, stream>>>(sum1b1, sq1b1, g1_0, be1_0, a1b1, c1b1, 64, (float)P1);

  lgk_mlp_kernel<32, 32, 64, false><<<B * S / 16, 256, 0, stream>>>(
      diff0, stdinv, w0_0, w0_1, a1b0, c1b0, nullptr, nullptr, sum2b0, sq2b0, nullptr, 0, 16);
  lgk_mlp_kernel<64, 64, 128, false><<<B * S / 16, 256, 0, stream>>>(
      diff1, stdinv + 16, w1_0, w1_1, a1b1, c1b1, nullptr, nullptr, sum2b1, sq2b1, nullptr, 64, 16);
  lgk_bnfin_kernel<<<1, 256, 0, stream>>>(sum2b0, sq2b0, g0_1, be0_1, a2b0, c2b0, 64, (float)P0);
  lgk_bnfin_kernel<<<1, 256, 0, stream>>>(sum2b1, sq2b1, g1_1, be1_1, a2b1, c2b1, 128, (float)P1);

  lgk_mlp_kernel<32, 32, 64, true><<<B * S / 16, 256, 0, stream>>>(
      diff0, stdinv, w0_0, w0_1, a1b0, c1b0, a2b0, c2b0, nullptr, nullptr, out1, 0, 16);
  lgk_mlp_kernel<64, 64, 128, true><<<B * S / 16, 256, 0, stream>>>(
      diff1, stdinv + 16, w1_0, w1_1, a1b1, c1b1, a2b1, c2b1, nullptr, nullptr, out1, 64, 16);
}